// BlockDense_89730456748629
// MI455X (gfx1250) — compile-verified
//
#include <hip/hip_runtime.h>

// Problem constants (from the reference)
#define GRP     32
#define WIN     128
#define HID     256
#define BATCH   4096
#define IN_COLS  (GRP * WIN)   // 4096
#define OUT_COLS (GRP * HID)   // 8192

// Tiling
#define MT 128                 // batch rows per block
#define NT 64                  // output cols per block (per group)
#define KT WIN                 // full K staged (128)
#define XS_STRIDE 130          // floats: 128 + 2 pad (TDM pad: 2 DWORDs / 128)
#define WS_STRIDE 72           // floats: 64 + 8 pad  (TDM pad: 8 DWORDs / 64)
                               // -> 2 k-rows = 144 ≡ 16 mod 64 banks: the two
                               //    half-wave B-frag loads hit disjoint banks
#define XS_FLOATS (MT * XS_STRIDE)          // 16640
#define WS_FLOATS (KT * WS_STRIDE)          // 9216
#define SMEM_FLOATS (XS_FLOATS + WS_FLOATS) // 25856 floats = 103424 B

typedef __attribute__((ext_vector_type(2))) float v2f;
typedef __attribute__((ext_vector_type(4))) float v4f;
typedef __attribute__((ext_vector_type(8))) float v8f;
typedef __attribute__((ext_vector_type(4))) unsigned int u32x4;
typedef __attribute__((ext_vector_type(4))) int i32x4;
typedef __attribute__((ext_vector_type(8))) int i32x8;

#if defined(__gfx1250__) && __has_builtin(__builtin_amdgcn_tensor_load_to_lds)
#define USE_TDM 1
#else
#define USE_TDM 0
#endif

#if USE_TDM
// Build a 2D-tile Tensor DMA Descriptor (D#) and issue TENSOR_LOAD_TO_LDS.
// Field packing per CDNA5 ISA §8.3/§8.4 (group0 128b, group1 256b; groups
// 2/3 zero for <=2D tensors). data_size=4B, pad_enable=1 (LDS-side padding).
__device__ __forceinline__ void tdm_load_2d(unsigned lds_byte_addr,
                                            const void* gptr,
                                            unsigned tensor_d0,
                                            unsigned tensor_d1,
                                            unsigned d0_stride_elems,
                                            unsigned tile_d0,
                                            unsigned tile_d1,
                                            unsigned pad_interval,  // 0=2,..,5=64,6=128 DW
                                            unsigned pad_amount)    // 0=1,..,7=8 DW
{
    unsigned long long ga = (unsigned long long)gptr;
    u32x4 g0 = { 1u,                                         // count=1 (valid user D#)
                 lds_byte_addr,                              // [63:32] lds_addr
                 (unsigned)(ga & 0xFFFFFFFFu),               // [95:64] global_addr lo
                 ((unsigned)(ga >> 32) & 0x01FFFFFFu)        // [120:96] global_addr hi
                   | (2u << 30) };                           // [127:126] type=2 (image)
    unsigned w0 = (2u << 16)                                 // data_size: 2 -> 4 bytes
                | (1u << 20)                                 // pad_enable
                | (pad_interval << 22)
                | (pad_amount << 25);
    i32x8 g1 = { (int)w0,
                 (int)((tensor_d0 & 0xFFFFu) << 16),         // abar_addr=0 | td0[15:0]
                 (int)((tensor_d0 >> 16) | ((tensor_d1 & 0xFFFFu) << 16)),
                 (int)((tensor_d1 >> 16) | (tile_d0 << 16)), // td1 hi | tile_dim0
                 (int)tile_d1,                               // tile_dim1 | tile_dim2=0
                 (int)d0_stride_elems,                       // tensor_dim0_stride[31:0]
                 0, 0 };                                     // stride hi | dim1_stride=0
    i32x4 gz4 = { 0, 0, 0, 0 };
#if __has_include(<hip/amd_detail/amd_gfx1250_TDM.h>)
    i32x8 gz8 = { 0, 0, 0, 0, 0, 0, 0, 0 };
    __builtin_amdgcn_tensor_load_to_lds(g0, g1, gz4, gz4, gz8, 0);   // clang-23 form
#else
    __builtin_amdgcn_tensor_load_to_lds(g0, g1, gz4, gz4, 0);        // ROCm 7.2 form
#endif
}
#endif // USE_TDM

__launch_bounds__(256, 2)
__global__ void block_dense_wmma_f32(const float* __restrict__ X,
                                     const float* __restrict__ W,
                                     float* __restrict__ Out)
{
    __shared__ __align__(16) float smem[SMEM_FLOATS];
    float* Xs = smem;               // [m][k], stride 130 floats
    float* Ws = smem + XS_FLOATS;   // [k][n], stride 72 floats

    const int nt = blockIdx.x;   // 0..3   : 64-col tile within the group's 256 cols
    const int mt = blockIdx.y;   // 0..31  : 128-row tile of the batch
    const int g  = blockIdx.z;   // 0..31  : group (diagonal block)

    const int tid     = threadIdx.x;
    const int lane    = tid & 31;
    const int wave    = tid >> 5;
    const int lane_lo = lane & 15;
    const int lane_hi = lane >> 4;

    const int m_base   = mt * MT;
    const int n_base   = nt * NT;
    const int gcol_in  = g * WIN;
    const int gcol_out = g * HID + n_base;

    const float* Xsrc = X + (size_t)m_base * IN_COLS + gcol_in;
    const float* Wsrc = W + (size_t)(g * WIN) * OUT_COLS + g * HID + n_base;

    // Real LDS byte addresses (low 32 bits of the generic pointer are the LDS
    // offset per the aperture mapping). The ptrtoint is also the *escape* that
    // stops LLVM from proving smem is never written by the TDM engine.
    const unsigned xs_lds = (unsigned)(unsigned long long)(uintptr_t)(void*)Xs;
    const unsigned ws_lds = (unsigned)(unsigned long long)(uintptr_t)(void*)Ws;

#if USE_TDM
    // One TDM DMA per tile; issued once (EXEC is ignored by tensor ops).
    if (wave == 0) {
        // X tile: 128 rows x 128 f32, LDS row stride 130 (pad 2 DW per 128 DW)
        tdm_load_2d(xs_lds, Xsrc, IN_COLS, BATCH, IN_COLS,
                    /*tile_d0=*/WIN, /*tile_d1=*/MT,
                    /*pad_interval=*/6u, /*pad_amount=*/1u);
        // W block: 128 k-rows x 64 f32, LDS row stride 72 (pad 8 DW per 64 DW)
        tdm_load_2d(ws_lds, Wsrc, OUT_COLS, IN_COLS, OUT_COLS,
                    /*tile_d0=*/NT, /*tile_d1=*/KT,
                    /*pad_interval=*/5u, /*pad_amount=*/7u);
        __builtin_amdgcn_s_wait_tensorcnt(0);
    }
#else
    // Fallback: manual staging (coalesced float4 loads -> padded LDS rows)
    {
        const int row_f4 = WIN / 4;
        for (int i = tid; i < MT * row_f4; i += 256) {
            const int r = i / row_f4, c4 = i % row_f4;
            v4f v = *(const v4f*)(Xsrc + (size_t)r * IN_COLS + c4 * 4);
            float* dst = &Xs[r * XS_STRIDE + c4 * 4];
            dst[0] = v.x; dst[1] = v.y; dst[2] = v.z; dst[3] = v.w;
        }
        const int row_f4w = NT / 4;
        for (int i = tid; i < KT * row_f4w; i += 256) {
            const int k = i / row_f4w, n4 = i % row_f4w;
            v4f v = *(const v4f*)(Wsrc + (size_t)k * OUT_COLS + n4 * 4);
            float* dst = &Ws[k * WS_STRIDE + n4 * 4];
            dst[0] = v.x; dst[1] = v.y; dst[2] = v.z; dst[3] = v.w;
        }
    }
#endif
    __syncthreads();

    // Compiler-level fence that may write escaped memory (smem escaped via the
    // ptrtoint above): forces real LDS reloads below, WITHOUT laundering the
    // pointers themselves — reads stay addrspace(3) and lower to ds_load_*.
    asm volatile("" : : "s"(xs_lds), "s"(ws_lds) : "memory");

    // ---- Compute: each wave -> 16 rows x 64 cols = 4 accum tiles of 16x16 f32 ----
    const int wrow = wave * 16;
    v8f acc[4] = {};

    // A frag (16x4 f32): lane holds row (wrow+lane_lo), K pair (k0 + 2*lane_hi)
    const float* a_row = Xs + (wrow + lane_lo) * XS_STRIDE + 2 * lane_hi;
    // B frag (4x16 f32): lane holds col (t*16+lane_lo); the two K values of its
    // pair live one k-row (WS_STRIDE) apart -> ds_load_2addr_b32 candidates.
    const float* b_col = Ws + lane_lo;

    #pragma unroll 4
    for (int k0 = 0; k0 < KT; k0 += 4) {
        const v2f a = *(const v2f*)(a_row + k0);
        const float* b_row = b_col + (k0 + 2 * lane_hi) * WS_STRIDE;
        #pragma unroll
        for (int t = 0; t < 4; ++t) {
            v2f b;
            b.x = b_row[t * 16];
            b.y = b_row[t * 16 + WS_STRIDE];
            acc[t] = __builtin_amdgcn_wmma_f32_16x16x4_f32(
                         /*neg_a=*/false, a, /*neg_b=*/false, b,
                         /*c_mod=*/(short)0, acc[t],
                         /*reuse_a=*/false, /*reuse_b=*/false);
        }
    }

    // ---- Store with ReLU ----
    // D layout: VGPR r -> M = r + 8*lane_hi ; N = lane_lo
    const size_t row0 = (size_t)(m_base + wrow + 8 * lane_hi);
    #pragma unroll
    for (int t = 0; t < 4; ++t) {
        const size_t col = (size_t)(gcol_out + t * 16 + lane_lo);
        #pragma unroll
        for (int r = 0; r < 8; ++r) {
            Out[(row0 + r) * OUT_COLS + col] = fmaxf(acc[t][r], 0.0f);
        }
    }
}

extern "C" void kernel_launch(void* const* d_in, const int* in_sizes, int n_in,
                              void* d_out, int out_size, void* d_ws, size_t ws_size,
                              hipStream_t stream) {
    (void)in_sizes; (void)n_in; (void)d_ws; (void)ws_size; (void)out_size;
    const float* X = (const float*)d_in[0];   // (4096, 4096) f32
    const float* W = (const float*)d_in[1];   // (4096, 8192) f32
    float* Out = (float*)d_out;               // (4096, 8192) f32

    dim3 grid(HID / NT, BATCH / MT, GRP);     // (4, 32, 32)
    dim3 block(256);
    block_dense_wmma_f32<<<grid, block, 0, stream>>>(X, W, Out);
}